// DiTEmb_dynamics_42202348651014
// MI455X (gfx1250) — compile-verified
//
#include <hip/hip_runtime.h>

typedef float v2f __attribute__((ext_vector_type(2)));
typedef float v8f __attribute__((ext_vector_type(8)));

#define BATCH    8
#define NNODE    256
#define XH_IN_C  9
#define XH_HID_C 64
#define POS_HID_C 192
#define MFREQ    64
#define ROWS     (BATCH * NNODE)   // 2048
#define OUT_COLS 256               // 64 + 192

// ---------------------------------------------------------------------------
// Kernel 1: masked mean removal + build padded A1 (2048 x 12) for GEMM1.
// One block per batch, 256 threads (one per node).  LDS float atomics
// (ds_add_f32) do the 4-way column reduction.
// ---------------------------------------------------------------------------
__global__ void prep_kernel(const float* __restrict__ xh,
                            const float* __restrict__ node_mask,
                            float* __restrict__ xc,      // (2048,4): cx,cy,cz,mask
                            float* __restrict__ a1,      // (2048,12) zero-padded
                            float* __restrict__ ninv) {  // (8,) 1/sum(mask)
  int b = blockIdx.x;
  int j = threadIdx.x;
  __shared__ float acc[4];
  if (j < 4) acc[j] = 0.0f;
  __syncthreads();

  int row = b * NNODE + j;
  const float* r = xh + (size_t)row * XH_IN_C;
  float m  = node_mask[row];
  float x0 = r[0], x1 = r[1], x2 = r[2];
  atomicAdd(&acc[0], x0 * m);
  atomicAdd(&acc[1], x1 * m);
  atomicAdd(&acc[2], x2 * m);
  atomicAdd(&acc[3], m);
  __syncthreads();

  float inv = 1.0f / acc[3];
  float c0 = (x0 - acc[0] * inv) * m;
  float c1 = (x1 - acc[1] * inv) * m;
  float c2 = (x2 - acc[2] * inv) * m;

  float4 xo; xo.x = c0; xo.y = c1; xo.z = c2; xo.w = m;
  ((float4*)xc)[row] = xo;

  float* ar = a1 + row * 12;
  ar[0] = c0; ar[1] = c1; ar[2] = c2;
  ar[3] = r[3]; ar[4] = r[4]; ar[5] = r[5];
  ar[6] = r[6]; ar[7] = r[7]; ar[8] = r[8];
  ar[9] = 0.0f; ar[10] = 0.0f; ar[11] = 0.0f;

  if (j == 0) ninv[b] = inv;
}

// ---------------------------------------------------------------------------
// Kernel 2: S[i,k] = (mask_i / N) * sum_j mask_j * {sin,cos}(2*pi*d_ij*f_k).
// One block per (b,i).  Threads: k = tid&63, j-partition = tid>>6.
// v_sin/v_cos hardware semantics are sin(2*pi*u), so u = d*freq with a
// period-1 fract reduction is exact and minimal.  1/N is folded in here so
// the downstream GEMM epilogue is branch-free.
// ---------------------------------------------------------------------------
__global__ void pe_accum_kernel(const float* __restrict__ xc,
                                const float* __restrict__ ninv,
                                float* __restrict__ S) {
  int row = blockIdx.x;          // b*256 + i
  int b   = row >> 8;
  int tid = threadIdx.x;         // 0..255

  __shared__ float d_sh[NNODE];
  __shared__ float m_sh[NNODE];
  __shared__ float red[512];     // [0:256) sin parts, [256:512) cos parts

  float4 xi = ((const float4*)xc)[row];
  float4 xj = ((const float4*)xc)[b * NNODE + tid];
  float dx = xi.x - xj.x, dy = xi.y - xj.y, dz = xi.z - xj.z;
  float sq = fmaxf(dx * dx + dy * dy + dz * dz, 0.0f);
  d_sh[tid] = sqrtf(sq + 1e-12f);
  m_sh[tid] = xj.w;
  __syncthreads();

  int k    = tid & (MFREQ - 1);
  int part = tid >> 6;
  // freq = 100^(k/64) = exp2(k * log2(100)/64)
  float freq = exp2f((float)k * 0.10381025296523030f);

  float sa = 0.0f, ca = 0.0f;
  int j0 = part * 64;
#pragma unroll 8
  for (int jj = 0; jj < 64; ++jj) {
    int   j  = j0 + jj;
    float u  = d_sh[j] * freq;       // angle in turns
    float uf = u - floorf(u);        // range-reduce for v_sin/v_cos
    float mj = m_sh[j];
    sa = fmaf(mj, __builtin_amdgcn_sinf(uf), sa);   // sin(2*pi*uf)
    ca = fmaf(mj, __builtin_amdgcn_cosf(uf), ca);   // cos(2*pi*uf)
  }
  red[(k << 2) | part]         = sa;
  red[256 + ((k << 2) | part)] = ca;
  __syncthreads();

  if (tid < 128) {
    int base = (tid < 64) ? (tid << 2) : (256 + ((tid - 64) << 2));
    float v = red[base] + red[base + 1] + red[base + 2] + red[base + 3];
    S[row * (2 * MFREQ) + tid] = v * xi.w * ninv[b];
  }
}

// ---------------------------------------------------------------------------
// Kernels 3/4: fp32 WMMA GEMM, one wave32 per 16x16 output tile.
// D = (A @ W) + bias, then * node_mask, written into out[:, colOff:].
// Fragment layouts per CDNA5 ISA 16x16x4 f32 tables: A lanes 0-15 hold
// K=k0,k0+1 of row lm, lanes 16-31 hold K=k0+2,k0+3; B mirrored by K;
// C/D vgpr r <-> M = r (+8 for high lane half), N = lane%16.
// ---------------------------------------------------------------------------
template <int KPAD, int KREAL>
__global__ __launch_bounds__(32) void wmma_gemm_kernel(
    const float* __restrict__ A, int lda,
    const float* __restrict__ W, int ldw,
    const float* __restrict__ bias,
    const float* __restrict__ xc,    // .w component holds per-row node mask
    float* __restrict__ out, int colOff) {
  int lane = threadIdx.x;            // wave32
  int half = lane >> 4;              // 0 -> lanes 0-15, 1 -> lanes 16-31
  int lm   = lane & 15;
  int row0 = blockIdx.x * 16;
  int col0 = blockIdx.y * 16;

  v8f c = {};
#pragma unroll
  for (int k0 = 0; k0 < KPAD; k0 += 4) {
    int ka = k0 + (half ? 2 : 0);
    // A fragment: two consecutive K values of row (row0+lm) -> one b64 load
    v2f a = *(const v2f*)&A[(row0 + lm) * lda + ka];
    // B fragment: same K split, column col0+lm; zero-pad beyond KREAL
    v2f bm;
    bm.x = (ka     < KREAL) ? W[(ka    ) * ldw + col0 + lm] : 0.0f;
    bm.y = (ka + 1 < KREAL) ? W[(ka + 1) * ldw + col0 + lm] : 0.0f;
    c = __builtin_amdgcn_wmma_f32_16x16x4_f32(false, a, false, bm,
                                              (short)0, c, false, false);
  }

  float bv = bias[col0 + lm];
#pragma unroll
  for (int r = 0; r < 8; ++r) {
    int   row  = row0 + r + (half ? 8 : 0);
    float mask = xc[row * 4 + 3];
    out[row * OUT_COLS + colOff + col0 + lm] = (c[r] + bv) * mask;
  }
}

// ---------------------------------------------------------------------------
// Launch.  Inputs (setup_inputs order): t, xh, node_mask, edge_mask,
// W_xh, b_xh, W_pos, b_pos.  t and edge_mask are unused by the reference.
// ---------------------------------------------------------------------------
extern "C" void kernel_launch(void* const* d_in, const int* in_sizes, int n_in,
                              void* d_out, int out_size, void* d_ws, size_t ws_size,
                              hipStream_t stream) {
  const float* xh        = (const float*)d_in[1];
  const float* node_mask = (const float*)d_in[2];
  const float* W_xh      = (const float*)d_in[4];
  const float* b_xh      = (const float*)d_in[5];
  const float* W_pos     = (const float*)d_in[6];
  const float* b_pos     = (const float*)d_in[7];
  float* out = (float*)d_out;

  float* ws   = (float*)d_ws;
  float* xc   = ws;                       // 2048*4  floats
  float* ninv = ws + ROWS * 4;            // 8 floats (padded to 64)
  float* a1   = ninv + 64;                // 2048*12 floats
  float* S    = a1 + ROWS * 12;           // 2048*128 floats  (~1.2 MB total)

  prep_kernel<<<BATCH, NNODE, 0, stream>>>(xh, node_mask, xc, a1, ninv);
  pe_accum_kernel<<<ROWS, NNODE, 0, stream>>>(xc, ninv, S);

  // xh_emb: (2048,9)@(9,64), K padded to 12 -> 3 wmma per tile
  wmma_gemm_kernel<12, 9><<<dim3(ROWS / 16, XH_HID_C / 16), 32, 0, stream>>>(
      a1, 12, W_xh, XH_HID_C, b_xh, xc, out, 0);

  // pe: (2048,128)@(128,192), 1/N already folded into S -> 32 wmma per tile
  wmma_gemm_kernel<128, 128><<<dim3(ROWS / 16, POS_HID_C / 16), 32, 0, stream>>>(
      S, 128, W_pos, POS_HID_C, b_pos, xc, out, XH_HID_C);
}